// ChannelAttention_19602230739500
// MI455X (gfx1250) — compile-verified
//
#include <hip/hip_runtime.h>

typedef float v2f __attribute__((ext_vector_type(2)));
typedef float v8f __attribute__((ext_vector_type(8)));

#define C_ 64
#define NLOG2 16            // N = 65536 = 1<<16

static __device__ __forceinline__ v8f wmma4(v2f a, v2f b, v8f c) {
  // D = A(16x4 f32) * B(4x16 f32) + C(16x16 f32)
  return __builtin_amdgcn_wmma_f32_16x16x4_f32(false, a, false, b, (short)0, c,
                                               false, false);
}

static __device__ __forceinline__ unsigned lds_addr(const void* p) {
  // low 32 bits of a generic pointer to __shared__ = LDS byte address
  return (unsigned)(unsigned long long)p;
}

static __device__ __forceinline__ void async_copy_b128(unsigned lds_byte_addr,
                                                       unsigned global_voff_bytes,
                                                       const float* sbase) {
  asm volatile("global_load_async_to_lds_b128 %0, %1, %2"
               :
               : "v"(lds_byte_addr), "v"(global_voff_bytes), "s"(sbase)
               : "memory");
}

// ---------------- zero the energy accumulator in ws ----------------
__global__ void zero_kernel(float* __restrict__ p, int n) {
  int i = blockIdx.x * blockDim.x + threadIdx.x;
  if (i < n) p[i] = 0.f;
}

// ---------------- phase 1: energy[b] = Q Q^T (split-K + WMMA f32) ----------------
#define P1_STR 68     // 64 + 4 pad -> bank-conflict-free ds_load_b64 frags
#define RED_STR 65    // 64 + 1 pad -> conflict-free ds_add_f32 reduction
__global__ __launch_bounds__(128) void gram_kernel(const float* __restrict__ x,
                                                   float* __restrict__ energy) {
  __shared__ float ldsQ[2][64 * P1_STR];   // double-buffered 64x64 tiles (34.8 KB)
  __shared__ float red[64 * RED_STR];      // 16.6 KB cross-wave reduction
  const int b = blockIdx.y;
  const int chunk = blockIdx.x;            // 32 chunks of 2048 columns
  const int tid = threadIdx.x;
  const int lane = tid & 31;
  const int wave = tid >> 5;
  const int lrow = lane & 15;
  const int lhalf = lane >> 4;

  const long xbase = ((long)b * C_) << NLOG2;
  const float* xc = x + xbase + chunk * 2048;

  // issue one 64x64 subtile (16 KB) as 8 async b128 copies per thread
  auto issue = [&](int sub, int buf) {
#pragma unroll
    for (int it = 0; it < 8; ++it) {
      int f = tid + it * 128;              // float4 index 0..1023
      int r = f >> 4;                      // 16 float4 per row
      int c4 = f & 15;
      unsigned voff = ((unsigned)r << (NLOG2 + 2)) + (unsigned)c4 * 16u;
      unsigned la = lds_addr(&ldsQ[buf][r * P1_STR + c4 * 4]);
      async_copy_b128(la, voff, xc + sub * 64);
    }
  };

  issue(0, 0);                             // prologue: start DMA early
  for (int i = tid; i < 64 * RED_STR; i += 128) red[i] = 0.f;

  v8f acc[4][4];
#pragma unroll
  for (int i = 0; i < 4; ++i)
#pragma unroll
    for (int j = 0; j < 4; ++j) acc[i][j] = v8f{};

  for (int sub = 0; sub < 32; ++sub) {
    const int cur = sub & 1;
    if (sub < 31) {
      issue(sub + 1, cur ^ 1);
      asm volatile("s_wait_asynccnt 0x8" ::: "memory");  // in-order: cur landed
    } else {
      asm volatile("s_wait_asynccnt 0x0" ::: "memory");
    }
    __syncthreads();                       // cur tile visible to all waves
    // wave owns 16 of the 64 columns -> 4 K-steps of 4
#pragma unroll
    for (int kk = 0; kk < 4; ++kk) {
      const int kb = wave * 16 + kk * 4 + 2 * lhalf;
      v2f frag[4];
#pragma unroll
      for (int i = 0; i < 4; ++i)
        frag[i] = *(const v2f*)&ldsQ[cur][(16 * i + lrow) * P1_STR + kb];
      // A-frag and B-frag of Q Q^T share the same layout: 4 loads feed 16 WMMAs
#pragma unroll
      for (int i = 0; i < 4; ++i)
#pragma unroll
        for (int j = 0; j < 4; ++j) acc[i][j] = wmma4(frag[i], frag[j], acc[i][j]);
    }
    __syncthreads();                       // all readers done before overwrite
  }

  // cross-wave reduction in LDS (ds_add_f32, conflict-free via pad 65)
#pragma unroll
  for (int i = 0; i < 4; ++i)
#pragma unroll
    for (int j = 0; j < 4; ++j)
#pragma unroll
      for (int v = 0; v < 8; ++v) {
        int c = 16 * i + v + 8 * lhalf;    // C/D layout: M = vgpr + 8*(lane/16)
        int d = 16 * j + lrow;             // N = lane%16
        atomicAdd(&red[c * RED_STR + d], acc[i][j][v]);
      }
  __syncthreads();
  float* eb = energy + b * 4096;
  for (int idx = tid; idx < 4096; idx += 128)
    atomicAdd(&eb[idx], red[(idx >> 6) * RED_STR + (idx & 63)]);
}

// ---------------- phase 2: attention = softmax(max(E)-E) = exp(min(E)-E)/sum ----------------
__global__ __launch_bounds__(32) void softmax_kernel(const float* __restrict__ energy,
                                                     float* __restrict__ attn) {
  const int row = blockIdx.x;              // 0..1023  (b*64 + c)
  const int lane = threadIdx.x;
  const float* e = energy + row * 64;
  float e0 = e[lane], e1 = e[lane + 32];
  float m = fminf(e0, e1);
#pragma unroll
  for (int off = 16; off > 0; off >>= 1) m = fminf(m, __shfl_xor(m, off, 32));
  float p0 = expf(m - e0), p1 = expf(m - e1);
  float s = p0 + p1;
#pragma unroll
  for (int off = 16; off > 0; off >>= 1) s += __shfl_xor(s, off, 32);
  float inv = 1.f / s;
  float* a = attn + row * 64;
  a[lane] = p0 * inv;
  a[lane + 32] = p1 * inv;
}

// ---------------- phase 3: out = gamma * (attn @ Q) + x ----------------
#define P3_ASTR 68    // 64 + 4 pad -> conflict-free ds_load_b64 A-frags
__global__ __launch_bounds__(128) void av_kernel(const float* __restrict__ x,
                                                 const float* __restrict__ attn,
                                                 const float* __restrict__ gamma,
                                                 float* __restrict__ out) {
  __shared__ float aLds[64 * P3_ASTR];
  const int b = blockIdx.y;
  const int tid = threadIdx.x;
  const int lane = tid & 31;
  const int wave = tid >> 5;
  const int lrow = lane & 15;
  const int lhalf = lane >> 4;

  // async preload of attention[b] (16 KB) into padded LDS
  const float* ab = attn + b * 4096;
#pragma unroll
  for (int it = 0; it < 8; ++it) {
    int f = tid + it * 128;                // float4 index 0..1023
    int c = f >> 4;                        // row (16 float4 per 64-wide row)
    int d4 = f & 15;
    unsigned voff = (unsigned)f * 16u;
    unsigned la = lds_addr(&aLds[c * P3_ASTR + d4 * 4]);
    async_copy_b128(la, voff, ab);
  }
  asm volatile("s_wait_asynccnt 0x0" ::: "memory");
  __syncthreads();

  const long xb = ((long)b * C_) << NLOG2;
  const int nbase = blockIdx.x * 256 + wave * 64;   // wave owns 64 columns

  v8f acc[4][4];
#pragma unroll
  for (int i = 0; i < 4; ++i)
#pragma unroll
    for (int j = 0; j < 4; ++j) acc[i][j] = v8f{};

  for (int k = 0; k < 16; ++k) {          // K = 64 channels, 4 per step
    const int kc = 4 * k + 2 * lhalf;
    v2f af[4];
#pragma unroll
    for (int i = 0; i < 4; ++i)
      af[i] = *(const v2f*)&aLds[(16 * i + lrow) * P3_ASTR + kc];
    v2f bf[4];
#pragma unroll
    for (int j = 0; j < 4; ++j) {
      const long col = nbase + 16 * j + lrow;
      bf[j][0] = x[xb + ((long)kc << NLOG2) + col];
      bf[j][1] = x[xb + ((long)(kc + 1) << NLOG2) + col];
    }
#pragma unroll
    for (int i = 0; i < 4; ++i)
#pragma unroll
      for (int j = 0; j < 4; ++j) acc[i][j] = wmma4(af[i], bf[j], acc[i][j]);
  }

  const float g = gamma[0];
#pragma unroll
  for (int i = 0; i < 4; ++i)
#pragma unroll
    for (int j = 0; j < 4; ++j)
#pragma unroll
      for (int v = 0; v < 8; ++v) {
        int c = 16 * i + v + 8 * lhalf;
        long off = xb + ((long)c << NLOG2) + nbase + 16 * j + lrow;
        out[off] = g * acc[i][j][v] + x[off];
      }
}

extern "C" void kernel_launch(void* const* d_in, const int* in_sizes, int n_in,
                              void* d_out, int out_size, void* d_ws, size_t ws_size,
                              hipStream_t stream) {
  const float* x = (const float*)d_in[0];
  const float* gamma = (const float*)d_in[1];
  float* out = (float*)d_out;
  float* energy = (float*)d_ws;             // 16*64*64 = 65536 floats (256 KB)
  float* attn = energy + 16 * 4096;         // another 256 KB

  zero_kernel<<<dim3(256), 256, 0, stream>>>(energy, 16 * 4096);
  gram_kernel<<<dim3(32, 16), 128, 0, stream>>>(x, energy);
  softmax_kernel<<<dim3(1024), 32, 0, stream>>>(energy, attn);
  av_kernel<<<dim3(256, 16), 128, 0, stream>>>(x, attn, gamma, out);
}